// CrowdDynamicNet_56410100466226
// MI455X (gfx1250) — compile-verified
//
#include <hip/hip_runtime.h>
#include <hip/hip_bf16.h>

typedef float v2f __attribute__((ext_vector_type(2)));
typedef float v8f __attribute__((ext_vector_type(8)));

static constexpr int kSEQ   = 2048;
static constexpr int kBATCH = 64;
static constexpr int kIN    = 512;
static constexpr int kH     = 3;
static constexpr int kL     = 4;
static constexpr int kG     = 3 * kH;             // 9 gate rows
static constexpr int kROWS  = kSEQ * kBATCH;      // 131072 GEMM rows

// ---------------------------------------------------------------------------
// Kernel A: gx0 = X(131072 x 512) @ W_ih0^T(512 x 9, padded to 16) + b_ih0
// Bandwidth-bound (256 MB of X); uses V_WMMA_F32_16X16X4_F32, one 16x16 tile
// of N per wave, 128 chained WMMAs along K. One wave owns a 16-row M tile,
// 8 waves per block -> 128 rows/block.
// ---------------------------------------------------------------------------
__global__ __launch_bounds__(256) void gru_in_proj_wmma(
    const float* __restrict__ X, const float* __restrict__ Wih0,
    const float* __restrict__ bih0, float* __restrict__ gx) {
  // Padded B panel in LDS: Bs[k][n], n in [0,16), zero for n >= 9.  32 KB.
  __shared__ float Bs[kIN * 16];
  const int tid = threadIdx.x;
  for (int idx = tid; idx < kIN * 16; idx += 256) {
    const int k = idx >> 4;
    const int n = idx & 15;
    Bs[idx] = (n < kG) ? Wih0[n * kIN + k] : 0.0f;
  }
  __syncthreads();

  const int wave = tid >> 5;
  const int lane = tid & 31;
  const int half = lane >> 4;   // which 16-lane half of the wave
  const int n16  = lane & 15;
  const int row0 = blockIdx.x * 128 + wave * 16;

  // A 16x4 f32 layout: lane (half,m=n16) holds K = kk + 2*half + {0,1} of row m.
  const float* arow = X + (size_t)(row0 + n16) * kIN + 2 * half;

  v8f c = {0.f, 0.f, 0.f, 0.f, 0.f, 0.f, 0.f, 0.f};
  for (int kk = 0; kk < kIN; kk += 4) {
    v2f a = *(const v2f*)(arow + kk);
    // B 4x16 f32 layout: VGPR i, half h -> K = kk + i + 2h, N = n16.
    v2f b;
    b.x = Bs[(kk + 2 * half) * 16 + n16];
    b.y = Bs[(kk + 2 * half + 1) * 16 + n16];
    c = __builtin_amdgcn_wmma_f32_16x16x4_f32(
        /*neg_a=*/false, a, /*neg_b=*/false, b,
        /*c_mod=*/(short)0, c, /*reuse_a=*/false, /*reuse_b=*/false);
  }

  // D layout: lane holds column N=n16 for M = v + 8*half, v in [0,8).
  if (n16 < kG) {
    const float bias = bih0[n16];
#pragma unroll
    for (int v = 0; v < 8; ++v) {
      const int m = v + 8 * half;
      gx[(size_t)(row0 + m) * kG + n16] = c[v] + bias;
    }
  }
}

// ---------------------------------------------------------------------------
// Kernel B: fused 4-layer GRU scan. 64 threads (one batch element each),
// weights broadcast from LDS, hidden state in registers. Latency-bound
// dependent chain (2048 x 4 steps) -- no matrix units useful at H=3.
// ---------------------------------------------------------------------------
__device__ __forceinline__ float sigmoid_f(float x) {
  return 1.0f / (1.0f + __expf(-x));
}
__device__ __forceinline__ float tanh_f(float x) {
  // stable: tanh(x) = sign(x) * (1 - 2/(exp(2|x|)+1)); exp overflow -> 1.
  const float e = __expf(2.0f * fabsf(x));
  return copysignf(1.0f - 2.0f / (e + 1.0f), x);
}

__device__ __forceinline__ void gru_cell(float h[kH], const float gxv[kG],
                                         const float* __restrict__ whh,
                                         const float* __restrict__ bhh) {
  float gh[kG];
#pragma unroll
  for (int g = 0; g < kG; ++g) {
    float s = bhh[g];
#pragma unroll
    for (int j = 0; j < kH; ++j) s += whh[g * kH + j] * h[j];
    gh[g] = s;
  }
#pragma unroll
  for (int j = 0; j < kH; ++j) {
    const float r = sigmoid_f(gxv[j] + gh[j]);
    const float z = sigmoid_f(gxv[kH + j] + gh[kH + j]);
    const float n = tanh_f(gxv[2 * kH + j] + r * gh[2 * kH + j]);
    h[j] = (1.0f - z) * n + z * h[j];
  }
}

__global__ __launch_bounds__(64) void gru_scan(
    const float* __restrict__ gx0, const float* __restrict__ hxs,
    const float* __restrict__ whh0, const float* __restrict__ bhh0,
    const float* __restrict__ wih_r, const float* __restrict__ whh_r,
    const float* __restrict__ bih_r, const float* __restrict__ bhh_r,
    float* __restrict__ out) {
  // LDS weight pack: [0..26]=whh0, [27..35]=bhh0,
  // then per layer l in 0..2 at 36+l*72: wih(27), whh(27), bih(9), bhh(9).
  __shared__ float Ws[36 + 3 * 72];
  const int tid = threadIdx.x;
  for (int i = tid; i < 27; i += 64) Ws[i] = whh0[i];
  for (int i = tid; i < 9; i += 64) Ws[27 + i] = bhh0[i];
  for (int i = tid; i < 3 * 27; i += 64)
    Ws[36 + (i / 27) * 72 + (i % 27)] = wih_r[i];
  for (int i = tid; i < 3 * 27; i += 64)
    Ws[36 + (i / 27) * 72 + 27 + (i % 27)] = whh_r[i];
  for (int i = tid; i < 3 * 9; i += 64)
    Ws[36 + (i / 9) * 72 + 54 + (i % 9)] = bih_r[i];
  for (int i = tid; i < 3 * 9; i += 64)
    Ws[36 + (i / 9) * 72 + 63 + (i % 9)] = bhh_r[i];
  __syncthreads();

  const int b = tid;
  float h[kL][kH];
#pragma unroll
  for (int l = 0; l < kL; ++l)
#pragma unroll
    for (int j = 0; j < kH; ++j) h[l][j] = hxs[(l * kBATCH + b) * kH + j];

  for (int t = 0; t < kSEQ; ++t) {
    float gxv[kG];
    const float* gp = gx0 + (size_t)(t * kBATCH + b) * kG;
#pragma unroll
    for (int g = 0; g < kG; ++g) gxv[g] = gp[g];
    gru_cell(h[0], gxv, &Ws[0], &Ws[27]);

#pragma unroll
    for (int l = 1; l < kL; ++l) {
      const float* wih = &Ws[36 + (l - 1) * 72];
      const float* whh = wih + 27;
      const float* bih = wih + 54;
      const float* bhh = wih + 63;
#pragma unroll
      for (int g = 0; g < kG; ++g) {
        float s = bih[g];
#pragma unroll
        for (int j = 0; j < kH; ++j) s += wih[g * kH + j] * h[l - 1][j];
        gxv[g] = s;
      }
      gru_cell(h[l], gxv, whh, bhh);
    }

#pragma unroll
    for (int j = 0; j < kH; ++j)
      out[(size_t)(t * kBATCH + b) * kH + j] = h[kL - 1][j];
  }

  // final hidden states, concatenated after the sequence output
  for (int l = 0; l < kL; ++l)
#pragma unroll
    for (int j = 0; j < kH; ++j)
      out[(size_t)kSEQ * kBATCH * kH + (l * kBATCH + b) * kH + j] = h[l][j];
}

// ---------------------------------------------------------------------------
extern "C" void kernel_launch(void* const* d_in, const int* in_sizes, int n_in,
                              void* d_out, int out_size, void* d_ws,
                              size_t ws_size, hipStream_t stream) {
  const float* x    = (const float*)d_in[0];
  const float* hxs  = (const float*)d_in[1];
  const float* wih0 = (const float*)d_in[2];
  const float* whh0 = (const float*)d_in[3];
  const float* bih0 = (const float*)d_in[4];
  const float* bhh0 = (const float*)d_in[5];
  const float* wihr = (const float*)d_in[6];
  const float* whhr = (const float*)d_in[7];
  const float* bihr = (const float*)d_in[8];
  const float* bhhr = (const float*)d_in[9];
  float* out = (float*)d_out;
  float* gx  = (float*)d_ws;  // kROWS * 9 floats = 4.5 MB scratch

  gru_in_proj_wmma<<<kROWS / 128, 256, 0, stream>>>(x, wih0, bih0, gx);
  gru_scan<<<1, 64, 0, stream>>>(gx, hxs, whh0, bhh0, wihr, whhr, bihr, bhhr,
                                 out);
}